// FreqDecouplingReconstruction_19877108646000
// MI455X (gfx1250) — compile-verified
//
#include <hip/hip_runtime.h>

#define Gn 8
#define K2c 9
#define GK2 72
#define Cc 128
#define Hh 192
#define Ww 192
#define Nn 4
#define HW (Hh * Ww)

typedef __attribute__((ext_vector_type(2))) float v2f;
typedef __attribute__((ext_vector_type(8))) float v8f;

// ---------------- Kernel 1: global average pool over HxW per (n,c) ----------------
__global__ __launch_bounds__(256)
void pool_kernel(const float* __restrict__ x, float* __restrict__ pooled) {
    int nc = blockIdx.x;                 // 0 .. N*C-1 (512)
    int tid = threadIdx.x;               // 256
    const float* base = x + (size_t)nc * HW;
    float s = 0.f;
    for (int i = tid; i < HW; i += 256) s += base[i];
    __shared__ float red[256];
    red[tid] = s;
    __syncthreads();
    for (int off = 128; off > 0; off >>= 1) {
        if (tid < off) red[tid] += red[tid + off];
        __syncthreads();
    }
    if (tid == 0) pooled[nc] = red[0] * (1.0f / (float)HW);
}

// ---------------- Kernel 2: filter generation via fp32 WMMA (one wave32) ----------------
// f = pooled @ w_conv.T        [4,128]x[128,72] -> padded to [16,80], K-steps of 4
// g = f @ w_gate.T             [4,72]x[72,72]   -> padded to [16,80], K = 72 exact
// f = f * sigmoid(g); BN(eval); softmax over 9 taps per (n,group) -> filt[4*72]
__global__ __launch_bounds__(32)
void filt_wmma_kernel(const float* __restrict__ pooled,
                      const float* __restrict__ w_conv,
                      const float* __restrict__ w_gate,
                      const float* __restrict__ bn_scale,
                      const float* __restrict__ bn_bias,
                      const float* __restrict__ bn_mean,
                      const float* __restrict__ bn_var,
                      float* __restrict__ filt) {
    __shared__ float fs[16][80];
    __shared__ float gs[16][80];
    __shared__ float hs[Nn][GK2];

    const int lane = threadIdx.x;        // 0..31
    const int m    = lane & 15;
    const int half = lane >> 4;
    const int ko   = half * 2;           // K offset within 4-chunk for vgpr0; +1 for vgpr1

    // A-row masking by multiply (keeps EXEC all-ones around WMMA)
    const int   mrow  = (m < Nn) ? m : 0;
    const float mmask = (m < Nn) ? 1.f : 0.f;

    v8f facc[5];
    #pragma unroll
    for (int t = 0; t < 5; ++t)
        #pragma unroll
        for (int i = 0; i < 8; ++i) facc[t][i] = 0.f;

    // ---- Stage 1: f = pooled @ w_conv.T ----
    for (int kb = 0; kb < Cc; kb += 4) {
        v2f a;
        a.x = pooled[mrow * Cc + kb + ko]     * mmask;
        a.y = pooled[mrow * Cc + kb + ko + 1] * mmask;
        #pragma unroll
        for (int t = 0; t < 5; ++t) {
            int   ncol = t * 16 + m;
            int   nn   = (ncol < GK2) ? ncol : 0;
            float bmk  = (ncol < GK2) ? 1.f : 0.f;
            v2f b;
            b.x = w_conv[nn * Cc + kb + ko]     * bmk;
            b.y = w_conv[nn * Cc + kb + ko + 1] * bmk;
            facc[t] = __builtin_amdgcn_wmma_f32_16x16x4_f32(
                false, a, false, b, (short)0, facc[t], false, false);
        }
    }
    // D layout: VGPR v, lanes 0-15 -> row v; lanes 16-31 -> row v+8; col = lane%16
    #pragma unroll
    for (int t = 0; t < 5; ++t)
        #pragma unroll
        for (int v = 0; v < 8; ++v)
            fs[v + half * 8][t * 16 + m] = facc[t][v];
    __syncthreads();

    // ---- Stage 2: g = f @ w_gate.T (A rows >=4 are already exactly zero) ----
    v8f gacc[5];
    #pragma unroll
    for (int t = 0; t < 5; ++t)
        #pragma unroll
        for (int i = 0; i < 8; ++i) gacc[t][i] = 0.f;

    for (int kb = 0; kb < GK2; kb += 4) {
        v2f a;
        a.x = fs[m][kb + ko];
        a.y = fs[m][kb + ko + 1];
        #pragma unroll
        for (int t = 0; t < 5; ++t) {
            int   ncol = t * 16 + m;
            int   nn   = (ncol < GK2) ? ncol : 0;
            float bmk  = (ncol < GK2) ? 1.f : 0.f;
            v2f b;
            b.x = w_gate[nn * GK2 + kb + ko]     * bmk;
            b.y = w_gate[nn * GK2 + kb + ko + 1] * bmk;
            gacc[t] = __builtin_amdgcn_wmma_f32_16x16x4_f32(
                false, a, false, b, (short)0, gacc[t], false, false);
        }
    }
    #pragma unroll
    for (int t = 0; t < 5; ++t)
        #pragma unroll
        for (int v = 0; v < 8; ++v)
            gs[v + half * 8][t * 16 + m] = gacc[t][v];
    __syncthreads();

    // ---- gating + BatchNorm (eval) ----
    for (int idx = lane; idx < Nn * GK2; idx += 32) {
        int n = idx / GK2, o = idx % GK2;
        float f = fs[n][o];
        float g = gs[n][o];
        float val = f * (1.f / (1.f + expf(-g)));
        val = (val - bn_mean[o]) * rsqrtf(bn_var[o] + 1e-5f) * bn_scale[o] + bn_bias[o];
        hs[n][o] = val;
    }
    __syncthreads();

    // ---- softmax over 9 taps; 4*8 = 32 groups == 32 lanes ----
    {
        int n = lane >> 3, grp = lane & 7;
        float mx = -3.4e38f;
        #pragma unroll
        for (int k = 0; k < K2c; ++k) mx = fmaxf(mx, hs[n][grp * K2c + k]);
        float e[K2c];
        float sum = 0.f;
        #pragma unroll
        for (int k = 0; k < K2c; ++k) {
            e[k] = expf(hs[n][grp * K2c + k] - mx);
            sum += e[k];
        }
        float inv = 1.f / sum;
        #pragma unroll
        for (int k = 0; k < K2c; ++k) filt[n * GK2 + grp * K2c + k] = e[k] * inv;
    }
}

// ---------------- Kernel 3: fused involution + hf reduction + reconstruction ----------------
// Block = 256 threads = 2 pixels x 128 channels. Both pixels share n (HW is even).
__global__ __launch_bounds__(256)
void main_kernel(const float* __restrict__ x, const float* __restrict__ raw,
                 const float* __restrict__ filt, const float* __restrict__ w_high,
                 const float* __restrict__ b_high, const float* __restrict__ gamma,
                 float* __restrict__ out) {
    __shared__ float filt_s[GK2];
    __shared__ float part[2][4];

    const int tid = threadIdx.x;
    const int p0  = blockIdx.x * 2;
    const int n   = p0 / HW;

    if (tid < GK2) filt_s[tid] = filt[n * GK2 + tid];
    __syncthreads();

    const int pix = tid >> 7;
    const int c   = tid & 127;
    const int rem = (p0 - n * HW) + pix;
    const int h   = rem / Ww;
    const int w   = rem % Ww;

    const float* xc = x + (size_t)(n * Cc + c) * HW;
    const int g9 = (c >> 4) * K2c;

    // 9-tap involution with reflect padding (taps hit L2: whole x < 192MB L2)
    float acc = 0.f;
    #pragma unroll
    for (int di = -1; di <= 1; ++di) {
        int hi = h + di;
        hi = (hi < 0) ? -hi : hi;
        hi = (hi >= Hh) ? (2 * Hh - 2 - hi) : hi;
        const float* row = xc + hi * Ww;
        #pragma unroll
        for (int dj = -1; dj <= 1; ++dj) {
            int wj = w + dj;
            wj = (wj < 0) ? -wj : wj;
            wj = (wj >= Ww) ? (2 * Ww - 2 - wj) : wj;
            acc += filt_s[g9 + (di + 1) * 3 + (dj + 1)] * row[wj];
        }
    }

    const int off = h * Ww + w;
    const float xv = xc[off];
    // per-pixel channel reduction of high * w_high  (wave32 shuffle + 4-wave LDS combine)
    float hv = (xv - acc) * w_high[c];
    #pragma unroll
    for (int d = 16; d > 0; d >>= 1) hv += __shfl_xor(hv, d, 32);
    if ((tid & 31) == 0) part[pix][(tid >> 5) & 3] = hv;
    __syncthreads();

    const float hf  = part[pix][0] + part[pix][1] + part[pix][2] + part[pix][3] + b_high[0];
    const float tau = expf(gamma[0]);            // D = 1.0
    const size_t gidx = (size_t)(n * Cc + c) * HW + off;
    const float rv = raw[gidx];
    // low - hf*tau, with low = low_part*raw + raw - low_part
    out[gidx] = acc * rv + rv - acc - hf * tau;
}

// ---------------- launcher ----------------
extern "C" void kernel_launch(void* const* d_in, const int* in_sizes, int n_in,
                              void* d_out, int out_size, void* d_ws, size_t ws_size,
                              hipStream_t stream) {
    const float* x        = (const float*)d_in[0];
    const float* raw      = (const float*)d_in[1];
    const float* w_conv   = (const float*)d_in[2];
    const float* w_gate   = (const float*)d_in[3];
    const float* bn_scale = (const float*)d_in[4];
    const float* bn_bias  = (const float*)d_in[5];
    const float* bn_mean  = (const float*)d_in[6];
    const float* bn_var   = (const float*)d_in[7];
    const float* w_high   = (const float*)d_in[8];
    const float* b_high   = (const float*)d_in[9];
    const float* gamma    = (const float*)d_in[10];
    float* out = (float*)d_out;

    float* pooled = (float*)d_ws;          // N*C = 512 floats
    float* filt   = pooled + Nn * Cc;      // N*G*K2 = 288 floats

    pool_kernel<<<dim3(Nn * Cc), dim3(256), 0, stream>>>(x, pooled);
    filt_wmma_kernel<<<dim3(1), dim3(32), 0, stream>>>(
        pooled, w_conv, w_gate, bn_scale, bn_bias, bn_mean, bn_var, filt);
    main_kernel<<<dim3(Nn * HW / 2), dim3(256), 0, stream>>>(
        x, raw, filt, w_high, b_high, gamma, out);
}